// AttentionGNNLayer_8701603741951
// MI455X (gfx1250) — compile-verified
//
#include <hip/hip_runtime.h>
#include <hip/hip_bf16.h>

typedef __attribute__((ext_vector_type(2))) float v2f;
typedef __attribute__((ext_vector_type(8))) float v8f;
typedef __attribute__((ext_vector_type(4))) unsigned int v4u;
typedef __attribute__((ext_vector_type(8))) int v8i;
typedef __attribute__((ext_vector_type(4))) int v4i;

// f32 WMMA: D(16x16) = A(16x4) * B(4x16) + C   -- full fp32 precision path
__device__ __forceinline__ v8f wmma4(v2f a, v2f b, v8f c) {
    return __builtin_amdgcn_wmma_f32_16x16x4_f32(
        /*neg_a=*/false, a, /*neg_b=*/false, b,
        /*c_mod=*/(short)0, c, /*reuse_a=*/false, /*reuse_b=*/false);
}

// Order-preserving float -> uint key (for atomic segment max incl. negatives)
__device__ __forceinline__ unsigned fkey(float f) {
    unsigned u = __float_as_uint(f);
    return (u & 0x80000000u) ? ~u : (u | 0x80000000u);
}

// ---------------------------------------------------------------------------
// TDM: load a [128 rows x 32 cols] f32 chunk of a row-major [128 x tensor_k]
// weight matrix into LDS. Pad feature inserts 4 dwords after every 32 dwords,
// giving a 36-dword LDS row stride => conflict-free B-fragment ds_load_b64.
// Descriptor layout per CDNA5 ISA ch.8 (group0 128b, group1 256b).
// This toolchain's builtin takes 6 args (g0,g1,g2,g3,g4,cpol); groups 2/3/4
// are zero for a 2-D tensor.
// ---------------------------------------------------------------------------
#define LDS_WSTRIDE 36
#define CHUNK_K     32

__device__ __forceinline__ void tdm_load_chunk(const float* gsrc,
                                               unsigned lds_byte,
                                               int tensor_k) {
    unsigned long long ga = (unsigned long long)(uintptr_t)gsrc;
    v4u g0;
    g0.x = 1u;                                   // count=1, no gather
    g0.y = lds_byte;                             // lds_addr [63:32]
    g0.z = (unsigned)(ga & 0xFFFFFFFFu);         // global_addr lo [95:64]
    g0.w = ((unsigned)(ga >> 32) & 0x01FFFFFFu)  // global_addr hi [120:96]
           | 0x80000000u;                        // type=2 ("image") [127:126]
    v8i g1;
    // data_size=4B(2), pad_enable, pad_interval=32dw(4), pad_amount=4dw(3)
    g1[0] = 0x07120000;
    g1[1] = (tensor_k & 0xFFFF) << 16;           // tensor_dim0 lo16 @ [63:48]
    g1[2] = ((tensor_k >> 16) & 0xFFFF)          // tensor_dim0 hi16 @ [79:64]
            | (128 << 16);                       // tensor_dim1 lo16 @ [95:80]
    g1[3] = (CHUNK_K << 16);                     // tile_dim0=32 @ [127:112]
    g1[4] = 128;                                 // tile_dim1=128 @ [143:128]
    g1[5] = tensor_k;                            // tensor_dim0_stride lo32
    g1[6] = 0;
    g1[7] = 0;
    v4i g2 = {0, 0, 0, 0};
    v4i g3 = {0, 0, 0, 0};
    v8i g4 = {0, 0, 0, 0, 0, 0, 0, 0};
    __builtin_amdgcn_tensor_load_to_lds(g0, g1, g2, g3, g4, 0);
}

// ---------------------------------------------------------------------------
// Workspace init
// ---------------------------------------------------------------------------
__global__ void init_ws_kernel(float* __restrict__ attn_agg,
                               float* __restrict__ mean_agg,
                               float* __restrict__ mean_cnt,
                               float* __restrict__ segsum,
                               unsigned* __restrict__ segmaxk,
                               long total, int N) {
    long i = (long)blockIdx.x * blockDim.x + threadIdx.x;
    if (i >= total) return;
    attn_agg[i] = 0.0f;
    mean_agg[i] = 0.0f;
    if (i < N) {
        mean_cnt[i] = 0.0f;
        segsum[i]   = 0.0f;
        segmaxk[i]  = 0x80000000u;   // fkey(0.0f): reference clamps seg max at >= 0
    }
}

// ---------------------------------------------------------------------------
// Edge attention logits:  e_ij = tanh([r_emb ‖ h_emb] @ Wattn^T + b) @ w0 + b0
// 8 waves/block, one 16-edge tile per wave. Weights TDM-staged in LDS,
// double-buffered across 8 K-chunks of 32.
// ---------------------------------------------------------------------------
__global__ void __launch_bounds__(256)
attn_logits_kernel(const float* __restrict__ node_emb,
                   const int*   __restrict__ er_src,
                   const int*   __restrict__ er_dst,
                   const float* __restrict__ Wattn,  // [128][256]
                   const float* __restrict__ battn,  // [128]
                   const float* __restrict__ w0,     // [128]
                   const float* __restrict__ w0b,    // [1]
                   float* __restrict__ e_out) {
    __shared__ float lds_w[2][128 * LDS_WSTRIDE];   // 2 x 18KB

    int tid   = threadIdx.x;
    int wave  = (int)(blockIdx.x * 8 + (tid >> 5)); // tile id (grid exact)
    int lane  = tid & 31;
    int m     = lane & 15;                          // A: M = lane&15
    int khalf = (lane >> 4) << 1;                   // upper half-wave: K+2,K+3
    long base = (long)wave * 16;

    int nd = er_dst[base + m];                      // r_emb -> cols [0,128)
    int ns = er_src[base + m];                      // h_emb -> cols [128,256)
    const float* rowd = node_emb + (long)nd * 128;
    const float* rows = node_emb + (long)ns * 128;

    v8f acc[8] = {};

    const int NCHUNK = 256 / CHUNK_K;               // 8
    if (tid < 32)
        tdm_load_chunk(Wattn, (unsigned)(uintptr_t)&lds_w[0][0], 256);

    for (int c = 0; c < NCHUNK; ++c) {
        if (tid < 32) {
            if (c + 1 < NCHUNK) {
                tdm_load_chunk(Wattn + (c + 1) * CHUNK_K,
                               (unsigned)(uintptr_t)&lds_w[(c + 1) & 1][0], 256);
                __builtin_amdgcn_s_wait_tensorcnt(1);   // chunk c landed
            } else {
                __builtin_amdgcn_s_wait_tensorcnt(0);
            }
        }
        __syncthreads();

        const float* wc = &lds_w[c & 1][0];
        int kcb = c * CHUNK_K;
#pragma unroll
        for (int k0 = 0; k0 < CHUNK_K; k0 += 4) {
            int kb = kcb + k0 + khalf;
            const float* ap = (kb < 128) ? (rowd + kb) : (rows + (kb - 128));
            v2f a; a.x = ap[0]; a.y = ap[1];
            int kl = k0 + khalf;
            v2f bfrag[8];
#pragma unroll
            for (int nb = 0; nb < 8; ++nb) {
                const float* bp = wc + (nb * 16 + m) * LDS_WSTRIDE + kl;
                bfrag[nb].x = bp[0]; bfrag[nb].y = bp[1];
            }
#pragma unroll
            for (int nb = 0; nb < 8; ++nb)
                acc[nb] = wmma4(a, bfrag[nb], acc[nb]);
        }
        __syncthreads();
    }

    // Fused epilogue: sum_n tanh(hidden + b[n]) * w0[n] per row
    float part[8];
#pragma unroll
    for (int v = 0; v < 8; ++v) part[v] = 0.0f;
#pragma unroll
    for (int nb = 0; nb < 8; ++nb) {
        int n = nb * 16 + m;
        float bn = battn[n], wn = w0[n];
#pragma unroll
        for (int v = 0; v < 8; ++v)
            part[v] += tanhf(acc[nb][v] + bn) * wn;
    }
    // reduce across the 16 lanes of each half-wave (D: M = v + 8*(lane>=16))
#pragma unroll
    for (int v = 0; v < 8; ++v) {
        float s = part[v];
        for (int off = 1; off < 16; off <<= 1) s += __shfl_xor(s, off, 32);
        part[v] = s;
    }
    if (m == 0) {
        float b0 = w0b[0];
        int moff = (lane >> 4) * 8;
#pragma unroll
        for (int v = 0; v < 8; ++v)
            e_out[base + moff + v] = part[v] + b0;
    }
}

// ---------------------------------------------------------------------------
// Segment softmax pieces
// ---------------------------------------------------------------------------
__global__ void segmax_kernel(const float* __restrict__ eij,
                              const int* __restrict__ src,
                              unsigned* __restrict__ segmaxk, long nE) {
    long e = (long)blockIdx.x * blockDim.x + threadIdx.x;
    if (e >= nE) return;
    atomicMax(segmaxk + src[e], fkey(eij[e]));
}

__global__ void expsum_kernel(float* __restrict__ eij,           // in: logits, out: exp
                              const int* __restrict__ src,
                              const unsigned* __restrict__ segmaxk,
                              float* __restrict__ segsum, long nE) {
    long e = (long)blockIdx.x * blockDim.x + threadIdx.x;
    if (e >= nE) return;
    int s = src[e];
    float mx = __uint_as_float(segmaxk[s] & 0x7fffffffu);  // max >= 0 by seeding
    float ev = expf(eij[e] - mx);
    eij[e] = ev;
    atomicAdd(segsum + s, ev);
}

__global__ void attn_scatter_kernel(const float* __restrict__ node_emb,
                                    const int* __restrict__ src,
                                    const int* __restrict__ dst,
                                    const float* __restrict__ expv,
                                    const float* __restrict__ segsum,
                                    float* __restrict__ attn_agg, long nE) {
    long idx = (long)blockIdx.x * blockDim.x + threadIdx.x;
    long e = idx >> 7;
    int  c = (int)(idx & 127);
    if (e >= nE) return;
    int s = src[e], d = dst[e];
    float attn = expv[e] / (segsum[s] + 1e-9f);
    atomicAdd(attn_agg + (long)s * 128 + c, attn * node_emb[(long)d * 128 + c]);
}

// mean_agg[seg] += w * node_emb[gat];  mean_cnt[seg] += w   (w==nullptr -> 1.0)
__global__ void mean_scatter_kernel(const float* __restrict__ node_emb,
                                    const int* __restrict__ seg,
                                    const int* __restrict__ gat,
                                    const float* __restrict__ w,
                                    float* __restrict__ mean_agg,
                                    float* __restrict__ mean_cnt, long nE) {
    long idx = (long)blockIdx.x * blockDim.x + threadIdx.x;
    long e = idx >> 7;
    int  c = (int)(idx & 127);
    if (e >= nE) return;
    int sg = seg[e], gt = gat[e];
    float wt = w ? w[e] : 1.0f;
    atomicAdd(mean_agg + (long)sg * 128 + c, wt * node_emb[(long)gt * 128 + c]);
    if (c == 0) atomicAdd(mean_cnt + sg, wt);
}

// ---------------------------------------------------------------------------
// Node update: out = tanh(emb@W1^T+b1) + tanh(attn_agg@W2^T+b2)
//                  + tanh((mean_agg/clamp(cnt,1))@W3^T+b3)
// 8 waves/block, one 16-node tile per wave. W1/W2/W3 TDM-staged in LDS as a
// flat double-buffered sequence of 12 K-chunks. Tail waves clamp their tile
// (EXEC stays all-ones for WMMA + barriers) and only guard the stores.
// ---------------------------------------------------------------------------
__global__ void __launch_bounds__(256)
node_update_kernel(const float* __restrict__ node_emb,
                   const float* __restrict__ attn_agg,
                   const float* __restrict__ mean_agg,
                   const float* __restrict__ mean_cnt,
                   const float* __restrict__ W1, const float* __restrict__ b1,
                   const float* __restrict__ W2, const float* __restrict__ b2,
                   const float* __restrict__ W3, const float* __restrict__ b3,
                   float* __restrict__ out, int num_tiles) {
    __shared__ float lds_w[2][128 * LDS_WSTRIDE];

    int tid  = threadIdx.x;
    int wave = (int)(blockIdx.x * 8 + (tid >> 5));
    bool valid = wave < num_tiles;
    int tile = valid ? wave : (num_tiles - 1);      // clamp: keep all waves live
    int lane  = tid & 31;
    int m     = lane & 15;
    int khalf = (lane >> 4) << 1;
    long base = (long)tile * 16;
    long node = base + m;

    const float* Arows[3] = { node_emb + node * 128,
                              attn_agg + node * 128,
                              mean_agg + node * 128 };
    const float* Ws[3] = { W1, W2, W3 };
    const float* bs[3] = { b1, b2, b3 };
    float invc = 1.0f / fmaxf(mean_cnt[node], 1.0f);

    v8f res[8] = {};

    const int NCHUNK = 12;                          // 3 weights x 4 K-chunks
    if (tid < 32)
        tdm_load_chunk(W1, (unsigned)(uintptr_t)&lds_w[0][0], 128);

    int i = 0;
    for (int p = 0; p < 3; ++p) {
        v8f acc[8] = {};
        float scale = (p == 2) ? invc : 1.0f;
        const float* Arow = Arows[p];
        for (int c = 0; c < 4; ++c, ++i) {
            if (tid < 32) {
                if (i + 1 < NCHUNK) {
                    int pn = (i + 1) >> 2, cn = (i + 1) & 3;
                    tdm_load_chunk(Ws[pn] + cn * CHUNK_K,
                                   (unsigned)(uintptr_t)&lds_w[(i + 1) & 1][0], 128);
                    __builtin_amdgcn_s_wait_tensorcnt(1);
                } else {
                    __builtin_amdgcn_s_wait_tensorcnt(0);
                }
            }
            __syncthreads();

            const float* wc = &lds_w[i & 1][0];
            int kcb = c * CHUNK_K;
#pragma unroll
            for (int k0 = 0; k0 < CHUNK_K; k0 += 4) {
                int kb = kcb + k0 + khalf;
                v2f a; a.x = Arow[kb] * scale; a.y = Arow[kb + 1] * scale;
                int kl = k0 + khalf;
                v2f bfrag[8];
#pragma unroll
                for (int nb = 0; nb < 8; ++nb) {
                    const float* bp = wc + (nb * 16 + m) * LDS_WSTRIDE + kl;
                    bfrag[nb].x = bp[0]; bfrag[nb].y = bp[1];
                }
#pragma unroll
                for (int nb = 0; nb < 8; ++nb)
                    acc[nb] = wmma4(a, bfrag[nb], acc[nb]);
            }
            __syncthreads();
        }
#pragma unroll
        for (int nb = 0; nb < 8; ++nb) {
            float bn = bs[p][nb * 16 + m];
#pragma unroll
            for (int v = 0; v < 8; ++v)
                res[nb][v] += tanhf(acc[nb][v] + bn);
        }
    }

    // D layout: row M = v + 8*(lane>=16), col = nb*16 + (lane&15)
    if (valid) {
        int moff = (lane >> 4) * 8;
#pragma unroll
        for (int nb = 0; nb < 8; ++nb)
#pragma unroll
            for (int v = 0; v < 8; ++v)
                out[(base + moff + v) * 128 + nb * 16 + m] = res[nb][v];
    }
}

// ---------------------------------------------------------------------------
extern "C" void kernel_launch(void* const* d_in, const int* in_sizes, int n_in,
                              void* d_out, int out_size, void* d_ws, size_t ws_size,
                              hipStream_t stream) {
    const float* node_emb  = (const float*)d_in[0];
    const int*   er_src    = (const int*)  d_in[1];
    const int*   er_dst    = (const int*)  d_in[2];
    const int*   ee_src    = (const int*)  d_in[3];
    const int*   ee_dst    = (const int*)  d_in[4];
    const float* ee_weight = (const float*)d_in[5];
    const int*   rr_src    = (const int*)  d_in[6];
    const int*   rr_dst    = (const int*)  d_in[7];
    const float* W_attn_w  = (const float*)d_in[8];   // [128][256]
    const float* W_attn_b  = (const float*)d_in[9];   // [128]
    const float* w0_w      = (const float*)d_in[10];  // [1][128]
    const float* w0_b      = (const float*)d_in[11];  // [1]
    const float* W1_w      = (const float*)d_in[12];
    const float* W1_b      = (const float*)d_in[13];
    const float* W2_w      = (const float*)d_in[14];
    const float* W2_b      = (const float*)d_in[15];
    const float* W3_w      = (const float*)d_in[16];
    const float* W3_b      = (const float*)d_in[17];
    float* out = (float*)d_out;

    const int  D    = 128;
    const int  N    = in_sizes[0] / D;
    const long E_ER = in_sizes[1];
    const long E_EE = in_sizes[3];
    const long E_RR = in_sizes[6];

    // Workspace layout
    float*    attn_agg = (float*)d_ws;                    // N*128
    float*    mean_agg = attn_agg + (long)N * D;          // N*128
    float*    mean_cnt = mean_agg + (long)N * D;          // N
    float*    segsum   = mean_cnt + N;                    // N
    unsigned* segmaxk  = (unsigned*)(segsum + N);         // N
    float*    eij      = (float*)(segmaxk + N);           // E_ER (logits -> exp in place)

    // 1) init accumulators
    {
        long total = (long)N * D;
        init_ws_kernel<<<(int)((total + 255) / 256), 256, 0, stream>>>(
            attn_agg, mean_agg, mean_cnt, segsum, segmaxk, total, N);
    }
    // 2) edge attention logits (WMMA + TDM). 800000/(16*8) = 6250 blocks exact
    {
        int tiles = (int)(E_ER / 16);
        attn_logits_kernel<<<tiles / 8, 256, 0, stream>>>(
            node_emb, er_src, er_dst, W_attn_w, W_attn_b, w0_w, w0_b, eij);
    }
    // 3) segment max
    segmax_kernel<<<(int)((E_ER + 255) / 256), 256, 0, stream>>>(eij, er_src, segmaxk, E_ER);
    // 4) exp + segment sum
    expsum_kernel<<<(int)((E_ER + 255) / 256), 256, 0, stream>>>(eij, er_src, segmaxk, segsum, E_ER);
    // 5) attention-weighted scatter of r_emb into attn_agg
    {
        long total = E_ER * D;
        attn_scatter_kernel<<<(int)((total + 255) / 256), 256, 0, stream>>>(
            node_emb, er_src, er_dst, eij, segsum, attn_agg, E_ER);
    }
    // 6) mean aggregation scatters: ER (dst<-src), EE (src<-w*dst), RR (src<-dst)
    {
        long total = E_ER * D;
        mean_scatter_kernel<<<(int)((total + 255) / 256), 256, 0, stream>>>(
            node_emb, er_dst, er_src, nullptr, mean_agg, mean_cnt, E_ER);
        total = E_EE * D;
        mean_scatter_kernel<<<(int)((total + 255) / 256), 256, 0, stream>>>(
            node_emb, ee_src, ee_dst, ee_weight, mean_agg, mean_cnt, E_EE);
        total = E_RR * D;
        mean_scatter_kernel<<<(int)((total + 255) / 256), 256, 0, stream>>>(
            node_emb, rr_src, rr_dst, nullptr, mean_agg, mean_cnt, E_RR);
    }
    // 7) node update GEMMs (WMMA + TDM). ceil(6250/8) = 782 blocks, tail clamped
    {
        int tiles = N / 16;
        node_update_kernel<<<(tiles + 7) / 8, 256, 0, stream>>>(
            node_emb, attn_agg, mean_agg, mean_cnt,
            W1_w, W1_b, W2_w, W2_b, W3_w, W3_b, out, tiles);
    }
}